// DCO_40450001994315
// MI455X (gfx1250) — compile-verified
//
#include <hip/hip_runtime.h>
#include <math.h>

// ---------------- constants ----------------
#define B_      64
#define P_      8
#define N0      32768          // samples per row after FIR (B x N0)
#define NN      131072         // upsampled length = FFT length = 2M-2
#define M_      65537          // spectrum bins
#define NTAPS   129
#define COEFF   0.3926990816987241f      // 2*pi*0.5*1e6/8e6 = pi/8
#define TWO_PI  6.2831853071795864f
#define CFO_SC  0.039269908169872414f    // 2*pi*5e4/8e6
#define INV_N0  (1.0f/32768.0f)
#define INV_NN  (1.0f/131072.0f)
#define Y_SIZE  4194304                   // 64*8*8192

typedef __attribute__((ext_vector_type(2))) float v2f;
typedef __attribute__((ext_vector_type(8))) float v8f;

__device__ __forceinline__ float2 cmul(float2 a, float c, float s) {
    return make_float2(c * a.x - s * a.y, c * a.y + s * a.x);
}

// ---------------- zero row sums ----------------
__global__ void zero_rowsum_k(float* rowsum) {
    int t = threadIdx.x;
    if (t < B_) rowsum[t] = 0.0f;
}

// ---------------- FIR as Toeplitz GEMM via V_WMMA_F32_16X16X4_F32 ----------------
// One wave (32 threads) computes 256 consecutive outputs of one row:
//   Out[m,n] (16x16, output index o = base + 16n + m)
//   A[m,k]  = h_scaled[k-m]            (16 x 144 banded, K padded to 144)
//   B[k,n]  = x[b, (base + 16n + k - 64) mod N0]
__global__ __launch_bounds__(32)
void fir_wmma_k(const float* __restrict__ x, const float* __restrict__ filt,
                float* __restrict__ xf_raw, float* __restrict__ rowsum) {
    __shared__ float hs[176];   // h_scaled with +16 offset, zero padded
    __shared__ float win[384];  // circular input window [base-64, base+320)

    const int lane = threadIdx.x;
    const int b    = blockIdx.y;
    const int base = blockIdx.x * 256;

    for (int i = lane; i < 176; i += 32) {
        int j = i - 16;
        hs[i] = (j >= 0 && j < NTAPS) ? filt[j] * COEFF : 0.0f;
    }
    const float* xrow = x + (size_t)b * N0;
    for (int i = lane; i < 384; i += 32) {
        int src = (base + i - 64) & (N0 - 1);   // circular pad
        win[i] = xrow[src];
    }
    __syncthreads();

    v8f acc = {};
    const int m    = lane & 15;             // A row / B column index
    const int koff = (lane >> 4) * 2;       // lanes 0-15: K={0,1}; lanes 16-31: K={2,3}

    for (int kt = 0; kt < 36; ++kt) {       // K = 144 = 36 * 4
        const int k0 = kt * 4 + koff;
        v2f Af, Bf;
        Af.x = hs[16 + k0     - m];
        Af.y = hs[16 + k0 + 1 - m];
        Bf.x = win[16 * m + k0];
        Bf.y = win[16 * m + k0 + 1];
        acc = __builtin_amdgcn_wmma_f32_16x16x4_f32(
                  false, Af, false, Bf, (short)0, acc, false, false);
    }

    // D layout: VGPR v, lanes 0-15 -> M=v, lanes 16-31 -> M=v+8; N = lane&15
    const int n     = lane & 15;
    const int mbase = (lane >> 4) * 8;
    float local = 0.0f;
    float* orow = xf_raw + (size_t)b * N0 + base;
#pragma unroll
    for (int v = 0; v < 8; ++v) {
        float val = acc[v];
        orow[16 * n + (mbase + v)] = val;
        local += val;
    }
    for (int off = 16; off > 0; off >>= 1) local += __shfl_down(local, off, 32);
    if (lane == 0) atomicAdd(&rowsum[b], local);
}

// ---------------- per-row exclusive prefix sum of (xf - mean) ----------------
__global__ __launch_bounds__(1024)
void row_scan_k(const float* __restrict__ xf_raw, const float* __restrict__ rowsum,
                float* __restrict__ S_ex) {
    __shared__ float part[1024];
    const int b = blockIdx.x;
    const int t = threadIdx.x;
    const float mean = rowsum[b] * INV_N0;
    const float* xr = xf_raw + (size_t)b * N0;
    float*       sr = S_ex   + (size_t)b * N0;

    float vals[32];
    float s = 0.0f;
    const int base = t * 32;
#pragma unroll
    for (int i = 0; i < 32; ++i) { vals[i] = xr[base + i] - mean; s += vals[i]; }
    part[t] = s;
    __syncthreads();
    for (int off = 1; off < 1024; off <<= 1) {      // Hillis-Steele inclusive scan
        float add = (t >= off) ? part[t - off] : 0.0f;
        __syncthreads();
        part[t] += add;
        __syncthreads();
    }
    float prefix = (t == 0) ? 0.0f : part[t - 1];   // exclusive across threads
#pragma unroll
    for (int i = 0; i < 32; ++i) { sr[base + i] = prefix; prefix += vals[i]; }
}

// ---------------- fused spectrum build + radix-2 first stage (Ls=1, w=1) ----------------
// out[2t] = spec(t) + spec(t+NN/2); out[2t+1] = spec(t) - spec(t+NN/2)
__global__ void fft_first_k(const float* __restrict__ pn_spec, const float* __restrict__ df,
                            const float* __restrict__ pn_real, const float* __restrict__ pn_imag,
                            float2* __restrict__ out) {
    const int idx = blockIdx.x * blockDim.x + threadIdx.x;   // over B_*NN/2
    const int b = idx >> 16;
    const int t = idx & (NN / 2 - 1);

    float2 v[2];
#pragma unroll
    for (int h = 0; h < 2; ++h) {
        const int k  = t + h * (NN / 2);
        const int km = (k < M_) ? k : (NN - k);              // mirror X[:,1:-1][::-1]
        float amp = 0.0f;
        if (k != 0) amp = (float)NN * sqrtf(df[km] * pn_spec[km]) * 0.70710678118654752f;
        const size_t src = (size_t)b * M_ + km;
        v[h] = make_float2(amp * pn_real[src], amp * pn_imag[src]);
    }
    float2* orow = out + (size_t)b * NN;
    orow[2 * t]     = make_float2(v[0].x + v[1].x, v[0].y + v[1].y);
    orow[2 * t + 1] = make_float2(v[0].x - v[1].x, v[0].y - v[1].y);
}

// ---------------- one Stockham radix-4 inverse-FFT stage (sign +i) ----------------
// Ls = 1<<s completed sub-size; thread t: k = t & (Ls-1), g = t >> s
// y_m = sum_j w_j*a_j*i^(mj), w_j = e^{+i*2pi*j*k/(4Ls)}
// out[g*4Ls + m*Ls + k] = y_m
__global__ void fft_radix4_k(const float2* __restrict__ in, float2* __restrict__ out, int s) {
    const int Ls  = 1 << s;
    const int idx = blockIdx.x * blockDim.x + threadIdx.x;   // over B_*NN/4
    const int b = idx >> 15;
    const int t = idx & (NN / 4 - 1);
    const int k = t & (Ls - 1);

    const float2* ir = in + (size_t)b * NN;
    const float2 a0 = ir[t];
    const float2 a1 = ir[t +     NN / 4];
    const float2 a2 = ir[t + 2 * NN / 4];
    const float2 a3 = ir[t + 3 * NN / 4];

    const float ang = (TWO_PI * 0.25f / (float)Ls) * (float)k;
    float s1, c1;
    __sincosf(ang, &s1, &c1);
    const float c2 = c1 * c1 - s1 * s1, s2 = 2.0f * c1 * s1;
    const float c3 = c1 * c2 - s1 * s2, s3 = c1 * s2 + s1 * c2;

    const float2 b1 = cmul(a1, c1, s1);
    const float2 b2 = cmul(a2, c2, s2);
    const float2 b3 = cmul(a3, c3, s3);

    const float2 e0 = make_float2(a0.x + b2.x, a0.y + b2.y);
    const float2 e1 = make_float2(a0.x - b2.x, a0.y - b2.y);
    const float2 o0 = make_float2(b1.x + b3.x, b1.y + b3.y);
    const float2 o1 = make_float2(b1.x - b3.x, b1.y - b3.y);

    const int j = ((t >> s) << (s + 2)) + k;
    float2* orow = out + (size_t)b * NN;
    orow[j]          = make_float2(e0.x + o0.x, e0.y + o0.y);   // y0 = e0 + o0
    orow[j +     Ls] = make_float2(e1.x - o1.y, e1.y + o1.x);   // y1 = e1 + i*o1
    orow[j + 2 * Ls] = make_float2(e0.x - o0.x, e0.y - o0.y);   // y2 = e0 - o0
    orow[j + 3 * Ls] = make_float2(e1.x + o1.y, e1.y - o1.x);   // y3 = e1 - i*o1
}

// ---------------- last radix-4 stage: write Re(y)/NN only (Ls=32768, g=0) ----------------
__global__ void fft_radix4_last_k(const float2* __restrict__ in, float* __restrict__ pn) {
    const int idx = blockIdx.x * blockDim.x + threadIdx.x;   // over B_*NN/4
    const int b = idx >> 15;
    const int k = idx & (NN / 4 - 1);                        // k = t, g = 0

    const float2* ir = in + (size_t)b * NN;
    const float2 a0 = ir[k];
    const float2 a1 = ir[k +     NN / 4];
    const float2 a2 = ir[k + 2 * NN / 4];
    const float2 a3 = ir[k + 3 * NN / 4];

    const float ang = (TWO_PI * 0.25f / (float)(NN / 4)) * (float)k;
    float s1, c1;
    __sincosf(ang, &s1, &c1);
    const float c2 = c1 * c1 - s1 * s1, s2 = 2.0f * c1 * s1;
    const float c3 = c1 * c2 - s1 * s2, s3 = c1 * s2 + s1 * c2;

    const float2 b1 = cmul(a1, c1, s1);
    const float2 b2 = cmul(a2, c2, s2);
    const float2 b3 = cmul(a3, c3, s3);

    const float e0x = a0.x + b2.x, e1x = a0.x - b2.x;
    const float o0x = b1.x + b3.x;
    const float o1y = b1.y - b3.y;

    float* prow = pn + (size_t)b * NN;
    prow[k]                = (e0x + o0x) * INV_NN;   // Re y0
    prow[k +     NN / 4]   = (e1x - o1y) * INV_NN;   // Re y1 = Re(e1 + i*o1)
    prow[k + 2 * NN / 4]   = (e0x - o0x) * INV_NN;   // Re y2
    prow[k + 3 * NN / 4]   = (e1x + o1y) * INV_NN;   // Re y3
}

// ---------------- fused gather / phase / sincos / pack ----------------
__global__ void finalize_k(const float* __restrict__ S_ex, const float* __restrict__ xf_raw,
                           const float* __restrict__ rowsum, const float* __restrict__ pn,
                           const float* __restrict__ rand_phase_u, const float* __restrict__ cfo_u,
                           const float* __restrict__ zero_u, const int* __restrict__ tio_raw,
                           float* __restrict__ out) {
    const int b = blockIdx.y;
    const int i = blockIdx.x * blockDim.x + threadIdx.x;     // 0..N0-1
    if (i >= N0) return;

    const float cfo = (cfo_u[b] * 2.0f - 1.0f) * CFO_SC;
    int tio = tio_raw[b] - 100;
    if (zero_u[b] < 0.5f) tio = 0;

    const int n = 4 * i + 3 - tio;                           // shift + take-every-4th(offset 3)
    float sv = 0.0f, cv = 0.0f;
    if (n >= 0 && n < NN) {
        const int q = n >> 2, r = n & 3;
        const float mean = rowsum[b] * INV_N0;
        const float ph = rand_phase_u[b] * TWO_PI
                       + cfo * (float)(n + 1)
                       + S_ex[(size_t)b * N0 + q]
                       + (float)(r + 1) * (xf_raw[(size_t)b * N0 + q] - mean) * 0.25f
                       + pn[(size_t)b * NN + n];
        __sincosf(ph, &sv, &cv);
    }
    const int p = i >> 12;            // /4096
    const int l = i & 4095;
    const size_t ybase = ((size_t)b * P_ + p) * 8192;
    out[ybase + l]        = sv;
    out[ybase + 4096 + l] = cv;
    if (i == 0) {
        out[Y_SIZE + b]      = cfo;          // cfo output [B,1]
        out[Y_SIZE + B_ + b] = (float)tio;   // tio output [B]
    }
}

// ---------------- launcher ----------------
extern "C" void kernel_launch(void* const* d_in, const int* in_sizes, int n_in,
                              void* d_out, int out_size, void* d_ws, size_t ws_size,
                              hipStream_t stream) {
    const float* x            = (const float*)d_in[0];
    const float* filt         = (const float*)d_in[1];
    const float* pn_spec      = (const float*)d_in[2];
    const float* df           = (const float*)d_in[3];
    const float* rand_phase_u = (const float*)d_in[4];
    const float* pn_real      = (const float*)d_in[5];
    const float* pn_imag      = (const float*)d_in[6];
    const float* cfo_u        = (const float*)d_in[7];
    const float* zero_u       = (const float*)d_in[8];
    const int*   tio_raw      = (const int*)d_in[9];
    float* out = (float*)d_out;

    char* ws = (char*)d_ws;
    float2* buf0   = (float2*)(ws);                         // 64 MB (later aliased by pn)
    float2* buf1   = (float2*)(ws + (size_t)67108864);      // 64 MB
    float*  pn     = (float*)(ws);                          // 32 MB, aliases dead buf0
    float*  xf_raw = (float*) (ws + (size_t)134217728);     // 8 MB
    float*  S_ex   = (float*) (ws + (size_t)142606336);     // 8 MB
    float*  rowsum = (float*) (ws + (size_t)150994944);     // 256 B

    zero_rowsum_k<<<dim3(1), dim3(64), 0, stream>>>(rowsum);
    fir_wmma_k<<<dim3(128, 64), dim3(32), 0, stream>>>(x, filt, xf_raw, rowsum);
    row_scan_k<<<dim3(64), dim3(1024), 0, stream>>>(xf_raw, rowsum, S_ex);

    // FFT: fused build+radix-2 (Ls: 1->2), then 7 radix-4 (Ls: 2 -> 32768),
    // then final radix-4 writing scaled real parts (Ls: 32768 -> 131072).
    fft_first_k<<<dim3(16384), dim3(256), 0, stream>>>(pn_spec, df, pn_real, pn_imag, buf0);
    int s = 1;
    for (int pass = 0; pass < 7; ++pass, s += 2) {
        const float2* fin  = (pass & 1) ? buf1 : buf0;
        float2*       fout = (pass & 1) ? buf0 : buf1;
        fft_radix4_k<<<dim3(8192), dim3(256), 0, stream>>>(fin, fout, s);
    }
    // after 7 passes the live data is in buf1 (pass 6: buf0 -> buf1); buf0 is dead.
    fft_radix4_last_k<<<dim3(8192), dim3(256), 0, stream>>>(buf1, pn);

    finalize_k<<<dim3(128, 64), dim3(256), 0, stream>>>(
        S_ex, xf_raw, rowsum, pn, rand_phase_u, cfo_u, zero_u, tio_raw, out);
}